// NPSCalculator_81037442941558
// MI455X (gfx1250) — compile-verified
//
#include <hip/hip_runtime.h>
#include <hip/hip_bf16.h>

typedef __attribute__((ext_vector_type(2))) float v2f;
typedef __attribute__((ext_vector_type(8))) float v8f;

#define EPS_F        1e-6f
#define NPIX_IMG     90000      // 300*300
#define CH_STRIDE    90000      // channel plane stride (elements)
#define IMG_STRIDE   270000     // 3*90000
#define TILES_IMG    5625       // 90000/16
#define N_COLORS     30
#define INV_SIZE     (1.0f / 4320000.0f)   // 16*3*300*300

// Raw gfx1250 min/max (mnemonics confirmed from disasm) — avoids the
// compiler's canonicalize (v_max v,v,v) before every llvm.minnum.
__device__ __forceinline__ float fmin3_asm(float a, float b, float c) {
    float d;
    asm("v_min3_num_f32 %0, %1, %2, %3" : "=v"(d) : "v"(a), "v"(b), "v"(c));
    return d;
}
__device__ __forceinline__ float fmin_asm(float a, float b) {
    float d;
    asm("v_min_num_f32 %0, %1, %2" : "=v"(d) : "v"(a), "v"(b));
    return d;
}
__device__ __forceinline__ float fmax0_asm(float a) {
    float d;
    asm("v_max_num_f32 %0, %1, 0" : "=v"(d) : "v"(a));
    return d;
}

// D[m=color][n=pixel] = A[16colors x 4] * B[4 x 16pixels] + C
//   A row m  = [-2e_r, -2e_g, -2e_b, 1],  e_i = c_i - EPS  (pad rows -> 0,0,0,1)
//   B col n  = [r, g, b, r^2+g^2+b^2]
//   C[m][*]  = e_r^2+e_g^2+e_b^2 + EPS   (per-row constant; 1e30 for pad rows)
// => D = sum((a-c+EPS)^2) + EPS.  Min over colors taken in squared space
// (sqrt monotonic) -> one v_sqrt_f32 per pixel.
__global__ void nps_wmma_kernel(const float* __restrict__ patch,
                                const float* __restrict__ pa,
                                float* __restrict__ out,
                                int numTiles)
{
    const int lane   = threadIdx.x & 31;
    const int waveIb = threadIdx.x >> 5;
    const int wavesPerBlock = blockDim.x >> 5;
    const int nWaves = gridDim.x * wavesPerBlock;
    const int n      = lane & 15;
    const bool hiHalf = (lane >= 16);

    // ---------------- palette -> A matrices + C accumulators (once) --------
    v2f a0, a1;     // A tile 0: colors 0..15 ; A tile 1: colors 16..31 (30,31 padded)
    v8f c0, c1;
    {
        const float er0 = pa[n * IMG_STRIDE + 0 * CH_STRIDE] - EPS_F;
        const float eg0 = pa[n * IMG_STRIDE + 1 * CH_STRIDE] - EPS_F;
        const float eb0 = pa[n * IMG_STRIDE + 2 * CH_STRIDE] - EPS_F;
        a0.x = hiHalf ? (-2.0f * eb0) : (-2.0f * er0);   // K2 | K0
        a0.y = hiHalf ? 1.0f          : (-2.0f * eg0);   // K3 | K1

        const int  p1    = 16 + n;
        const bool valid = (p1 < N_COLORS);
        const int  p1c   = valid ? p1 : 0;
        const float er1 = pa[p1c * IMG_STRIDE + 0 * CH_STRIDE] - EPS_F;
        const float eg1 = pa[p1c * IMG_STRIDE + 1 * CH_STRIDE] - EPS_F;
        const float eb1 = pa[p1c * IMG_STRIDE + 2 * CH_STRIDE] - EPS_F;
        a1.x = hiHalf ? (valid ? -2.0f * eb1 : 0.0f) : (valid ? -2.0f * er1 : 0.0f);
        a1.y = hiHalf ? 1.0f                         : (valid ? -2.0f * eg1 : 0.0f);

        const int rbase = hiHalf ? 8 : 0;
        #pragma unroll
        for (int i = 0; i < 8; ++i) {
            const int m0 = rbase + i;
            const float fr0 = pa[m0 * IMG_STRIDE + 0 * CH_STRIDE] - EPS_F;
            const float fg0 = pa[m0 * IMG_STRIDE + 1 * CH_STRIDE] - EPS_F;
            const float fb0 = pa[m0 * IMG_STRIDE + 2 * CH_STRIDE] - EPS_F;
            c0[i] = fr0 * fr0 + fg0 * fg0 + fb0 * fb0 + EPS_F;

            const int  m1  = 16 + rbase + i;
            const bool v1  = (m1 < N_COLORS);
            const int  m1c = v1 ? m1 : 0;
            const float fr1 = pa[m1c * IMG_STRIDE + 0 * CH_STRIDE] - EPS_F;
            const float fg1 = pa[m1c * IMG_STRIDE + 1 * CH_STRIDE] - EPS_F;
            const float fb1 = pa[m1c * IMG_STRIDE + 2 * CH_STRIDE] - EPS_F;
            c1[i] = v1 ? (fr1 * fr1 + fg1 * fg1 + fb1 * fb1 + EPS_F) : 1e30f;
        }
    }

    // hoisted cross-half bpermute address: ((lane ^ 16) & 31) * 4, constant
    const int xHalfAddr = ((lane ^ 16) & 31) << 2;

    // ---------------- grid-stride over 16-pixel tiles -----------------------
    // tile index is wave-uniform: force scalar, maintain (tile-in-image,
    // element-offset) incrementally -> no per-tile integer division, and the
    // loop branch is a scalar branch (EXEC stays all-ones for WMMA).
    int tile = __builtin_amdgcn_readfirstlane(blockIdx.x * wavesPerBlock + waveIb);
    int tpi  = tile % TILES_IMG;
    int elem = (tile / TILES_IMG) * IMG_STRIDE + tpi * 16;
    const int tileStep = nWaves;
    const int elemStep = nWaves * 16;

    float acc = 0.0f;
    while (tile < numTiles) {
        const int base = elem + n;

        const float r  = patch[base];
        const float g  = patch[base + CH_STRIDE];
        const float bl = patch[base + 2 * CH_STRIDE];
        const float ss = r * r + g * g + bl * bl;

        v2f b;                                   // B 4x16: lanes 0-15: K0,K1; 16-31: K2,K3
        b.x = hiHalf ? bl : r;                   // K2=b  | K0=r
        b.y = hiHalf ? ss : g;                   // K3=ss | K1=g

        v8f d0 = __builtin_amdgcn_wmma_f32_16x16x4_f32(
            false, a0, false, b, (short)0, c0, false, false);
        v8f d1 = __builtin_amdgcn_wmma_f32_16x16x4_f32(
            false, a1, false, b, (short)0, c1, false, false);

        // min over this half's 16 colors: 8-instruction v_min3 tree
        const float t0 = fmin3_asm(d0[0], d0[1], d0[2]);
        const float t1 = fmin3_asm(d0[3], d0[4], d0[5]);
        const float t2 = fmin3_asm(d0[6], d0[7], d1[0]);
        const float t3 = fmin3_asm(d1[1], d1[2], d1[3]);
        const float t4 = fmin3_asm(d1[4], d1[5], d1[6]);
        const float u0 = fmin3_asm(t0, t1, t2);
        const float u1 = fmin3_asm(t3, t4, d1[7]);
        float mn = fmin_asm(u0, u1);
        // merge the other half's colors: raw ds_bpermute with hoisted address
        const float other = __int_as_float(
            __builtin_amdgcn_ds_bpermute(xHalfAddr, __float_as_int(mn)));
        mn = fmin_asm(mn, other);
        // single raw-HW sqrt per pixel
        acc += __builtin_amdgcn_sqrtf(fmax0_asm(mn));

        tile += tileStep;
        tpi  += tileStep;
        elem += elemStep;
        if (tpi >= TILES_IMG) { tpi -= TILES_IMG; elem += IMG_STRIDE - NPIX_IMG; }
        if (tpi >= TILES_IMG) { tpi -= TILES_IMG; elem += IMG_STRIDE - NPIX_IMG; }
    }

    // both half-waves hold identical pixel values -> wave sum is exactly 2x
    #pragma unroll
    for (int s = 16; s >= 1; s >>= 1) {
        const int a = ((lane ^ s) & 31) << 2;
        acc += __int_as_float(__builtin_amdgcn_ds_bpermute(a, __float_as_int(acc)));
    }

    __shared__ float sp[32];
    if (lane == 0) sp[waveIb] = acc * 0.5f;      // undo half-wave duplication (exact)
    __syncthreads();
    if (threadIdx.x == 0) {
        float s = 0.0f;
        for (int w = 0; w < wavesPerBlock; ++w) s += sp[w];
        atomicAdd(out, s * INV_SIZE);
    }
}

extern "C" void kernel_launch(void* const* d_in, const int* in_sizes, int n_in,
                              void* d_out, int out_size, void* d_ws, size_t ws_size,
                              hipStream_t stream) {
    (void)n_in; (void)d_ws; (void)ws_size; (void)out_size;
    const float* patch = (const float*)d_in[0];   // [16,3,300,300] f32
    const float* pa    = (const float*)d_in[1];   // [30,3,300,300] f32 (constant over HxW)
    float* out = (float*)d_out;

    const int numPix   = in_sizes[0] / 3;         // 1,440,000
    const int numTiles = numPix / 16;             // 90,000

    hipMemsetAsync(out, 0, sizeof(float), stream);

    const int block = 256;                        // 8 waves per block
    const int grid  = 1024;                       // 8192 waves, ~11 tiles each
    nps_wmma_kernel<<<grid, block, 0, stream>>>(patch, pa, out, numTiles);
}